// LSTMLayer_31129922962190
// MI455X (gfx1250) — compile-verified
//
#include <hip/hip_runtime.h>

// LSTM: T=1024, B=64, I=512, H=512. f32 throughout (matches reference precision).
// Persistent kernel: 32 WGs x 128 threads (4 waves). WG p owns hidden columns
// [16p,16p+16); wave w owns batch rows [16w,16w+16) for all 4 gate groups.
// W slab (4 groups x 16 cols x K=1024 f32 = 256KB) lives in LDS for the whole run.
// c-state lives in registers in WMMA C-tile layout. h history flows through d_out.
// Device-wide barrier per timestep via atomic counter in d_ws.

#define TT 1024
#define BB 64
#define II 512
#define HH 512
#define GG (4 * HH)
#define NWG 32
#define NTHREADS 128
#define KTOT (II + HH) // 1024

typedef __attribute__((ext_vector_type(2))) float v2f;
typedef __attribute__((ext_vector_type(8))) float v8f;

__device__ __forceinline__ float sigm_f(float x) {
  return 1.0f / (1.0f + __expf(-x));
}
__device__ __forceinline__ float tanh_f(float x) {
  return 2.0f / (1.0f + __expf(-2.0f * x)) - 1.0f;
}

__global__ void __launch_bounds__(NTHREADS, 1)
lstm_persistent(const float* __restrict__ input,  // [T,B,I]
                const float* __restrict__ h0,     // [B,H]
                const float* __restrict__ c0,     // [B,H]
                const float* __restrict__ W_ih,   // [4H,I]
                const float* __restrict__ W_hh,   // [4H,H]
                const float* __restrict__ b_ih,   // [4H]
                const float* __restrict__ b_hh,   // [4H]
                float* __restrict__ out,          // [T,B,H] ++ h_f[B,H] ++ c_f[B,H]
                unsigned* __restrict__ barrier_cnt) {
  // Packed W slab in LDS: pair index = ((grp*512 + k/2)*16 + n), floats at *2+(k&1).
  // B-fragment for k-chunk k0: lane reads pair (k0/2 + laneHalf) for column n=lane&15.
  extern __shared__ float smem[]; // 4*512*16*2 floats = 256 KB

  const int tid  = threadIdx.x;
  const int wg   = blockIdx.x;        // 0..31 -> hidden cols [16*wg, 16*wg+16)
  const int wave = tid >> 5;          // 0..3  -> batch rows  [16*wave, ...)
  const int lane = tid & 31;
  const int nlane = lane & 15;        // N / M-low index within tile
  const int ksel  = lane >> 4;        // lane half selects k-pair
  const int j0 = wg * 16;
  const int r0 = wave * 16;

  // ---- One-time LDS fill of the weight slab (coalesced over k) ----
  for (int gn = 0; gn < 64; ++gn) {
    const int grp = gn >> 4, n = gn & 15;
    const int grow = grp * HH + j0 + n; // gate row in [0,2048)
    const float* wih = W_ih + (size_t)grow * II;
    const float* whh = W_hh + (size_t)grow * HH;
    for (int k = tid; k < KTOT; k += NTHREADS) {
      const float v = (k < II) ? wih[k] : whh[k - II];
      smem[((grp * 512 + (k >> 1)) * 16 + n) * 2 + (k & 1)] = v;
    }
  }
  __syncthreads();
  const v2f* __restrict__ Wlds2 = (const v2f*)smem;

  // ---- Per-lane constants ----
  const int col = j0 + nlane;           // hidden column this lane's tile slot maps to
  float bsum[4];
#pragma unroll
  for (int g = 0; g < 4; ++g)
    bsum[g] = b_ih[g * HH + col] + b_hh[g * HH + col];

  // C-tile layout: VGPR v, lanes 0-15 -> M=v ; lanes 16-31 -> M=8+v.
  const int bRow = r0 + 8 * ksel;       // + v gives batch row
  float c_r[8];
  float h_l[8];
#pragma unroll
  for (int v = 0; v < 8; ++v)
    c_r[v] = c0[(size_t)(bRow + v) * HH + col];

  const int arow = r0 + nlane;          // A-matrix row (M) this lane loads

  // ---- Recurrent loop ----
  for (int t = 0; t < TT; ++t) {
    v8f acc[4] = {};

    // K-part 1: x_t (K = 0..511)
    const float* __restrict__ Xt = input + (size_t)t * BB * II;
    const float* __restrict__ Xrow = Xt + (size_t)arow * II;
#pragma unroll 4
    for (int kc = 0; kc < II / 4; ++kc) {
      const int k0 = kc * 4;
      const v2f a = *(const v2f*)(Xrow + k0 + 2 * ksel);
      const int kp = (k0 >> 1) + ksel;
#pragma unroll
      for (int g = 0; g < 4; ++g) {
        const v2f b = Wlds2[(g * 512 + kp) * 16 + nlane];
        acc[g] = __builtin_amdgcn_wmma_f32_16x16x4_f32(
            false, a, false, b, (short)0, acc[g], false, false);
      }
    }

    // K-part 2: h_{t-1} (K = 512..1023)
    const float* __restrict__ Hp =
        (t == 0) ? h0 : (out + (size_t)(t - 1) * BB * HH);
    const float* __restrict__ Hrow = Hp + (size_t)arow * HH;
#pragma unroll 4
    for (int kc = 0; kc < HH / 4; ++kc) {
      const int k0 = kc * 4;
      const v2f a = *(const v2f*)(Hrow + k0 + 2 * ksel);
      const int kp = ((k0 + II) >> 1) + ksel;
#pragma unroll
      for (int g = 0; g < 4; ++g) {
        const v2f b = Wlds2[(g * 512 + kp) * 16 + nlane];
        acc[g] = __builtin_amdgcn_wmma_f32_16x16x4_f32(
            false, a, false, b, (short)0, acc[g], false, false);
      }
    }

    // ---- Elementwise LSTM cell update (fully wave-local) ----
    float* __restrict__ Hout = out + (size_t)t * BB * HH;
#pragma unroll
    for (int v = 0; v < 8; ++v) {
      const float ig = sigm_f(acc[0][v] + bsum[0]);
      const float fg = sigm_f(acc[1][v] + bsum[1]);
      const float gg = tanh_f(acc[2][v] + bsum[2]);
      const float og = sigm_f(acc[3][v] + bsum[3]);
      const float cn = fg * c_r[v] + ig * gg;
      const float hn = og * tanh_f(cn);
      c_r[v] = cn;
      h_l[v] = hn;
      Hout[(size_t)(bRow + v) * HH + col] = hn;
    }

    // ---- Device-wide barrier: make h_t visible before anyone starts t+1 ----
    __threadfence(); // release: each wave drains its own h stores to device scope
    __syncthreads();
    if (tid == 0) {
      __hip_atomic_fetch_add(barrier_cnt, 1u, __ATOMIC_RELEASE,
                             __HIP_MEMORY_SCOPE_AGENT);
      const unsigned target = (unsigned)(t + 1) * NWG;
      while (__hip_atomic_load(barrier_cnt, __ATOMIC_ACQUIRE,
                               __HIP_MEMORY_SCOPE_AGENT) < target) {
        __builtin_amdgcn_s_sleep(2);
      }
      __threadfence(); // acquire: invalidate WGP-local cache before h_t reads
    }
    __syncthreads();
  }

  // ---- Final state outputs ----
  float* __restrict__ hf = out + (size_t)TT * BB * HH;
  float* __restrict__ cf = hf + BB * HH;
#pragma unroll
  for (int v = 0; v < 8; ++v) {
    hf[(size_t)(bRow + v) * HH + col] = h_l[v];
    cf[(size_t)(bRow + v) * HH + col] = c_r[v];
  }
}

extern "C" void kernel_launch(void* const* d_in, const int* in_sizes, int n_in,
                              void* d_out, int out_size, void* d_ws,
                              size_t ws_size, hipStream_t stream) {
  (void)in_sizes; (void)n_in; (void)out_size; (void)ws_size;
  const float* input = (const float*)d_in[0];
  const float* h0    = (const float*)d_in[1];
  const float* c0    = (const float*)d_in[2];
  const float* W_ih  = (const float*)d_in[3];
  const float* W_hh  = (const float*)d_in[4];
  const float* b_ih  = (const float*)d_in[5];
  const float* b_hh  = (const float*)d_in[6];
  float* out = (float*)d_out;
  unsigned* cnt = (unsigned*)d_ws;

  // Zero the grid-barrier counter (capture-legal).
  hipMemsetAsync(d_ws, 0, 256, stream);

  const size_t ldsBytes = (size_t)4 * 512 * 16 * 2 * sizeof(float); // 256 KB
  // Allow >default dynamic LDS (idempotent; ignore error if unsupported).
  (void)hipFuncSetAttribute((const void*)lstm_persistent,
                            hipFuncAttributeMaxDynamicSharedMemorySize,
                            (int)ldsBytes);

  lstm_persistent<<<dim3(NWG), dim3(NTHREADS), ldsBytes, stream>>>(
      input, h0, c0, W_ih, W_hh, b_ih, b_hh, out, cnt);
}